// MLP_44272522887405
// MI455X (gfx1250) — compile-verified
//
#include <hip/hip_runtime.h>
#include <math.h>

typedef __attribute__((ext_vector_type(2))) float v2f;
typedef __attribute__((ext_vector_type(8))) float v8f;

#define WAVES_PER_BLOCK 3
#define BLOCK_THREADS   (WAVES_PER_BLOCK * 32)
#define TILES_PER_WAVE  8
#define ROWS_PER_TILE   16
#define XT_STRIDE       132   // padded floats per X-tile row (bank-conflict-free WMMA-A reads)
#define H1_STRIDE       20

#define W1T_FLOATS (32 * 64)            // [step s][lane l][j] -> s*64 + 2*l + j
#define XT_FLOATS  (ROWS_PER_TILE * XT_STRIDE)
#define H1_FLOATS  (ROWS_PER_TILE * H1_STRIDE)

// ---- gfx1250 async global->LDS copy (ASYNCcnt-tracked, no VGPR staging) ----
// GVS mode: mem = SGPR_base + VGPR_offset ; VDST vgpr holds LDS byte address.
__device__ __forceinline__ void async_copy_b128(unsigned lds_byte_addr,
                                                unsigned g_byte_off,
                                                const float* g_base) {
    asm volatile("global_load_async_to_lds_b128 %0, %1, %2"
                 :: "v"(lds_byte_addr), "v"(g_byte_off), "s"(g_base)
                 : "memory");
}
__device__ __forceinline__ void wait_asynccnt(int n) {
    if (n == 0)
        asm volatile("s_wait_asynccnt 0x0" ::: "memory");
    else
        asm volatile("s_wait_asynccnt 0x10" ::: "memory");  // <= 16 outstanding
}

__global__ __launch_bounds__(BLOCK_THREADS)
void mlp_fused_kernel(const float* __restrict__ X,
                      const float* __restrict__ W1, const float* __restrict__ B1,
                      const float* __restrict__ W2, const float* __restrict__ B2,
                      const float* __restrict__ W3, const float* __restrict__ B3,
                      float* __restrict__ out, int batch)
{
    __shared__ float sW1T[W1T_FLOATS];                    // 8 KB: W1^T in WMMA-B lane layout
    __shared__ float sB1[16];
    __shared__ float sW2[80];
    __shared__ float sB2[8];
    __shared__ float sW3[8];
    __shared__ float sB3v[1];
    __shared__ float sX[WAVES_PER_BLOCK][2][XT_FLOATS];   // per-wave double-buffered X tile
    __shared__ float sH1[WAVES_PER_BLOCK][H1_FLOATS];     // per-wave H1 tile (16x10, padded)

    const int tid  = threadIdx.x;
    const int lane = tid & 31;
    const int wid  = tid >> 5;
    const int nn   = lane & 15;            // N index (B/C/D lane mapping)
    const int koff = (lane >> 4) << 1;     // K offset within 4-wide step: 0 or 2

    // ---- stage weights once per block; W1^T pre-swizzled into WMMA-B lane layout ----
    // B fragment for step s: vgpr j, lane l holds W1T[k = 4s + 2*(l>=16) + j][n = l%16]
    for (int idx = tid; idx < W1T_FLOATS; idx += BLOCK_THREADS) {
        int s = idx >> 6;
        int r = idx & 63;
        int l = r >> 1;
        int j = r & 1;
        int n = l & 15;
        int k = 4 * s + ((l >> 4) << 1) + j;
        sW1T[idx] = (n < 10) ? W1[n * 128 + k] : 0.0f;    // zero-pad N 10..15
    }
    if (tid < 16) sB1[tid] = (tid < 10) ? B1[tid] : 0.0f;
    if (tid < 80) sW2[tid] = W2[tid];
    if (tid < 8)  sB2[tid] = B2[tid];
    if (tid < 8)  sW3[tid] = W3[tid];
    if (tid == 0) sB3v[0] = B3[0];
    __syncthreads();

    const long long gwave = (long long)blockIdx.x * WAVES_PER_BLOCK + wid;
    const long long tile0 = gwave * TILES_PER_WAVE;
    float* h1 = sH1[wid];

    // Per-lane invariant pieces of the async copy addressing.
    // Tile copy: 16 issues of b128/lane; issue `it` covers row it, float4 col = lane.
    const unsigned lds_lane_off = (unsigned)(lane * 16);        // bytes within LDS row
    const unsigned g_lane_off   = (unsigned)(lane * 16);        // bytes within 512B row

    // ---- prologue: kick off async copy of tile 0 into buffer 0 ----
    {
        const long long row0 = tile0 * ROWS_PER_TILE;
        if (row0 < batch) {
            const float* gb = X + row0 * 128;
            unsigned lb = (unsigned)(unsigned long long)(&sX[wid][0][0]);
            #pragma unroll
            for (int it = 0; it < 16; ++it)
                async_copy_b128(lb + (unsigned)(it * XT_STRIDE * 4) + lds_lane_off,
                                (unsigned)(it * 512) + g_lane_off, gb);
        }
    }

    for (int t = 0; t < TILES_PER_WAVE; ++t) {
        const long long row0 = (tile0 + t) * ROWS_PER_TILE;
        if (row0 >= batch) break;

        // ---- issue async copy of tile t+1 into the other buffer ----
        const bool have_next = (t + 1 < TILES_PER_WAVE) && (row0 + ROWS_PER_TILE < batch);
        if (have_next) {
            const float* gb = X + (row0 + ROWS_PER_TILE) * 128;
            unsigned lb = (unsigned)(unsigned long long)(&sX[wid][(t + 1) & 1][0]);
            #pragma unroll
            for (int it = 0; it < 16; ++it)
                async_copy_b128(lb + (unsigned)(it * XT_STRIDE * 4) + lds_lane_off,
                                (unsigned)(it * 512) + g_lane_off, gb);
        }
        // prefetch tile t+2 toward L2 (global_prefetch_b8)
        if (t + 2 < TILES_PER_WAVE && row0 + 2 * ROWS_PER_TILE < batch) {
            const float* nx = X + (row0 + 2 * ROWS_PER_TILE) * 128;
            __builtin_prefetch(nx + lane * 32, 0, 1);
            __builtin_prefetch(nx + 1024 + lane * 32, 0, 1);
        }

        // ---- wait for tile t's 16 async loads (they complete in order) ----
        wait_asynccnt(have_next ? 16 : 0);

        // ---- layer 1: C(16x16) = Xtile(16x128) @ W1^T(128x16) via WMMA ----
        // Two independent accumulator chains halve the serial C->C depth.
        const float* xt = &sX[wid][t & 1][0];
        v8f acc0 = {0.f, 0.f, 0.f, 0.f, 0.f, 0.f, 0.f, 0.f};
        v8f acc1 = {0.f, 0.f, 0.f, 0.f, 0.f, 0.f, 0.f, 0.f};
        const float* arow = &xt[nn * XT_STRIDE + koff];     // A: row M=nn, K stripe
        #pragma unroll
        for (int s = 0; s < 32; s += 2) {
            v2f a0 = *(const v2f*)(arow + 4 * s);                 // ds_load_b64, all-64-bank
            v2f b0 = *(const v2f*)&sW1T[s * 64 + 2 * lane];       // ds_load_b64, conflict-free
            v2f a1 = *(const v2f*)(arow + 4 * s + 4);
            v2f b1 = *(const v2f*)&sW1T[(s + 1) * 64 + 2 * lane];
            acc0 = __builtin_amdgcn_wmma_f32_16x16x4_f32(
                false, a0, false, b0, (short)0, acc0, false, false);
            acc1 = __builtin_amdgcn_wmma_f32_16x16x4_f32(
                false, a1, false, b1, (short)0, acc1, false, false);
        }

        // ---- bias + ReLU; transpose C layout (M in vgpr/half, N in lane) to LDS rows ----
        if (nn < 10) {
            const int mbase = (lane >> 4) << 3;             // 0 or 8
            const float bb = sB1[nn];
            #pragma unroll
            for (int v = 0; v < 8; ++v) {
                float h = (acc0[v] + acc1[v]) + bb;
                h1[(mbase + v) * H1_STRIDE + nn] = fmaxf(h, 0.0f);
            }
        }

        // ---- layers 2+3: one row per lane (tiny: ~100 FLOPs/row) ----
        if (lane < 16) {
            const float* hr = &h1[lane * H1_STRIDE];
            float z = sB3v[0];
            #pragma unroll
            for (int j = 0; j < 8; ++j) {
                float s2 = sB2[j];
                #pragma unroll
                for (int i = 0; i < 10; ++i)
                    s2 = fmaf(hr[i], sW2[j * 10 + i], s2);
                z = fmaf(fmaxf(s2, 0.0f), sW3[j], z);
            }
            if (row0 + lane < batch)
                out[row0 + lane] = 1.0f / (1.0f + __expf(-z));
        }
        // EXEC reconverges here before next iteration's WMMAs
    }
}

extern "C" void kernel_launch(void* const* d_in, const int* in_sizes, int n_in,
                              void* d_out, int out_size, void* d_ws, size_t ws_size,
                              hipStream_t stream) {
    const float* X  = (const float*)d_in[0];
    const float* W1 = (const float*)d_in[1];
    const float* B1 = (const float*)d_in[2];
    const float* W2 = (const float*)d_in[3];
    const float* B2 = (const float*)d_in[4];
    const float* W3 = (const float*)d_in[5];
    const float* B3 = (const float*)d_in[6];
    float* out = (float*)d_out;

    const int batch = out_size;  // output is [B, 1] f32
    const long long tiles  = (batch + ROWS_PER_TILE - 1) / ROWS_PER_TILE;
    const long long waves  = (tiles + TILES_PER_WAVE - 1) / TILES_PER_WAVE;
    const int blocks = (int)((waves + WAVES_PER_BLOCK - 1) / WAVES_PER_BLOCK);

    mlp_fused_kernel<<<blocks, BLOCK_THREADS, 0, stream>>>(
        X, W1, B1, W2, B2, W3, B3, out, batch);
}